// FreqSS2D_2370821947981
// MI455X (gfx1250) — compile-verified
//
#include <hip/hip_runtime.h>
#include <hip/hip_bf16.h>

// ---------------------------------------------------------------------------
// SS2D (VMamba) forward for MI455X / gfx1250.
// GEMMs use V_WMMA_F32_16X16X4_F32 (full fp32 precision, wave32 tile = 16x16).
// Selective scan uses a 3-phase chunked parallel scan (critical path 4096->~128).
// ---------------------------------------------------------------------------

namespace {
constexpr int BN  = 2;
constexpr int HH  = 64;
constexpr int WW  = 64;
constexpr int LL  = HH * WW;        // 4096
constexpr int DM  = 96;             // D_MODEL
constexpr int DI  = 192;            // D_INNER
constexpr int NS  = 16;             // D_STATE
constexpr int RK  = 6;              // DT_RANK
constexpr int KG  = 4;              // scan directions
constexpr int CD  = RK + 2 * NS;    // 38 rows of x_dbl
constexpr int NCH = 64;             // chunks along L
constexpr int CL  = LL / NCH;       // 64 steps per chunk
}

typedef float v2f __attribute__((ext_vector_type(2)));
typedef float v8f __attribute__((ext_vector_type(8)));

// scan-order index l of direction k -> spatial hw index into xconv
__device__ __forceinline__ int sp_map(int k, int l) {
  int j = (k & 2) ? (LL - 1 - l) : l;      // directions 2,3 are flipped
  if (k & 1) j = (j % HH) * WW + (j / HH); // directions 1,3 scan in wh order
  return j;
}

// One wave computes a 16x16 fp32 tile of A(MxK) * B(KxN)^T-style product where
// both operands are row-major with K contiguous. Lane layout per CDNA5 ISA:
//   A 16x4 : lane%16 = M row; VGPR0 = K{0,2}, VGPR1 = K{1,3} (half = lane/16)
//   B 4x16 : lane%16 = N col; same K-pair split across halves
//   C/D    : VGPR r -> row r + 8*(lane/16), col = lane%16
__device__ __forceinline__ v8f wave_tile_gemm(const float* __restrict__ arow,
                                              const float* __restrict__ brow,
                                              int K, int koff, bool bvalid) {
  v8f c = {0.f, 0.f, 0.f, 0.f, 0.f, 0.f, 0.f, 0.f};
  for (int kk = 0; kk < K; kk += 4) {
    v2f a, b;
    a.x = arow[kk + koff];
    a.y = arow[kk + koff + 1];
    b.x = bvalid ? brow[kk + koff] : 0.f;
    b.y = bvalid ? brow[kk + koff + 1] : 0.f;
    c = __builtin_amdgcn_wmma_f32_16x16x4_f32(false, a, false, b, (short)0, c,
                                              false, false);
  }
  return c;
}

// ---- 1) in_proj: xz = x @ in_proj_w^T ; split into xp (conv branch) and z ---
__global__ void k_inproj(const float* __restrict__ x,   // (B*L, 96)
                         const float* __restrict__ w,   // (384, 96)
                         float* __restrict__ xp,        // (B,L,192)
                         float* __restrict__ zb) {      // (B,L,192)
  const int lane = threadIdx.x;
  const int half = lane >> 4, t16 = lane & 15;
  const int koff = 2 * half;
  const int m0 = blockIdx.x * 16;
  const int n0 = blockIdx.y * 16;
  const float* arow = x + (size_t)(m0 + t16) * DM;
  const float* brow = w + (size_t)(n0 + t16) * DM;
  v8f c = wave_tile_gemm(arow, brow, DM, koff, true);
  const int col = n0 + t16;
#pragma unroll
  for (int r = 0; r < 8; ++r) {
    int g = m0 + r + 8 * half;
    int b = g / LL, l = g % LL;
    float v = c[r];
    if (col < DI) xp[((size_t)b * LL + l) * DI + col] = v;
    else          zb[((size_t)b * LL + l) * DI + (col - DI)] = v;
  }
}

// ---- 2) depthwise 3x3 conv (SAME, cross-correlation) + bias + SiLU ---------
__global__ void k_conv_silu(const float* __restrict__ xp,
                            const float* __restrict__ cw,  // (192,1,3,3)
                            const float* __restrict__ cb,
                            float* __restrict__ xc) {      // (B,L,192)
  int idx = blockIdx.x * blockDim.x + threadIdx.x;
  if (idx >= BN * LL * DI) return;
  int d = idx % DI;
  int m = (idx / DI) % LL;
  int b = idx / (DI * LL);
  int h = m / WW, w = m % WW;
  float acc = cb[d];
#pragma unroll
  for (int i = 0; i < 3; ++i) {
    int h2 = h + i - 1;
    if (h2 < 0 || h2 >= HH) continue;
#pragma unroll
    for (int j = 0; j < 3; ++j) {
      int w2 = w + j - 1;
      if (w2 < 0 || w2 >= WW) continue;
      acc += xp[((size_t)b * LL + h2 * WW + w2) * DI + d] * cw[d * 9 + i * 3 + j];
    }
  }
  xc[idx] = acc / (1.f + __expf(-acc));   // SiLU
}

// ---- 3) x_dbl[b,k,l,c] = sum_d xs[b,k,d,l] * x_proj_weight[k,c,d] ----------
__global__ void k_xdbl(const float* __restrict__ xc,
                       const float* __restrict__ xpw,   // (4,38,192)
                       float* __restrict__ xdbl) {      // (B,K,L,38)
  const int lane = threadIdx.x;
  const int half = lane >> 4, t16 = lane & 15;
  const int koff = 2 * half;
  const int m0 = blockIdx.x * 16;
  const int n0 = blockIdx.y * 16;
  const int bk = blockIdx.z;
  const int b = bk >> 2, k = bk & 3;
  const int l = m0 + t16;
  const float* arow = xc + ((size_t)b * LL + sp_map(k, l)) * DI;  // gather dir
  const int col = n0 + t16;
  const bool bvalid = col < CD;                 // pad N 38 -> 48
  const float* brow = xpw + ((size_t)k * CD + (bvalid ? col : 0)) * DI;
  v8f c = wave_tile_gemm(arow, brow, DI, koff, bvalid);
  if (col < CD) {
#pragma unroll
    for (int r = 0; r < 8; ++r) {
      int ll = m0 + r + 8 * half;
      xdbl[(((size_t)b * KG + k) * LL + ll) * CD + col] = c[r];
    }
  }
}

// ---- 4) delta = softplus(dt_proj(x_dbl[:RK]) + dt_bias) --------------------
__global__ void k_delta(const float* __restrict__ xdbl,
                        const float* __restrict__ dtw,  // (4,192,6)
                        const float* __restrict__ dtb,  // (4,192)
                        float* __restrict__ delta) {    // (B,K,L,192)
  int idx = blockIdx.x * blockDim.x + threadIdx.x;
  if (idx >= BN * KG * LL * DI) return;
  int d = idx % DI;
  int l = (idx / DI) % LL;
  int k = (idx / (DI * LL)) % KG;
  int b = idx / (DI * LL * KG);
  const float* row  = xdbl + (((size_t)b * KG + k) * LL + l) * CD;
  const float* wrow = dtw + ((size_t)k * DI + d) * RK;
  float acc = dtb[k * DI + d];
#pragma unroll
  for (int r = 0; r < RK; ++r) acc += row[r] * wrow[r];
  delta[idx] = fmaxf(acc, 0.f) + log1pf(__expf(-fabsf(acc)));  // stable softplus
}

// ---- 5) scan phase 1: per chunk, local scan from h=0 -> q[16]; S = sum(dt) -
// chunk transition is exp(A_n * S) analytically (diagonal A), so no P needed.
__global__ void k_scan_pass1(const float* __restrict__ delta,
                             const float* __restrict__ xc,
                             const float* __restrict__ xdbl,
                             const float* __restrict__ alog,
                             float* __restrict__ qb,   // (B,K,NCH,192,16)
                             float* __restrict__ sb) { // (B,K,NCH,192)
  const int d = threadIdx.x;
  const int cid = blockIdx.x;
  const int c = cid % NCH, k = (cid / NCH) % KG, b = cid / (NCH * KG);
  float A[NS], h[NS];
#pragma unroll
  for (int n = 0; n < NS; ++n) {
    A[n] = -__expf(alog[((size_t)k * DI + d) * NS + n]);
    h[n] = 0.f;
  }
  float S = 0.f;
  const size_t dbase  = ((size_t)b * KG + k) * LL * DI + d;
  const size_t xdbase = ((size_t)b * KG + k) * LL * CD;
  for (int t = 0; t < CL; ++t) {
    const int l = c * CL + t;
    const float dt = delta[dbase + (size_t)l * DI];
    const float u  = xc[((size_t)b * LL + sp_map(k, l)) * DI + d];
    S += dt;
    const float du = dt * u;
    const float* Brow = xdbl + xdbase + (size_t)l * CD + RK;
#pragma unroll
    for (int n = 0; n < NS; ++n) h[n] = h[n] * __expf(A[n] * dt) + du * Brow[n];
  }
  const size_t off = (((size_t)b * KG + k) * NCH + c) * DI + d;
#pragma unroll
  for (int n = 0; n < NS; ++n) qb[off * NS + n] = h[n];
  sb[off] = S;
}

// ---- 6) scan phase 2: serial combine over the 64 chunks -> h_start ---------
__global__ void k_scan_pass2(const float* __restrict__ qb,
                             const float* __restrict__ sb,
                             const float* __restrict__ alog,
                             float* __restrict__ hs) { // (B,K,NCH,192,16)
  const int d = threadIdx.x;
  const int bk = blockIdx.x;
  const int k = bk % KG, b = bk / KG;
  float A[NS], h[NS];
#pragma unroll
  for (int n = 0; n < NS; ++n) {
    A[n] = -__expf(alog[((size_t)k * DI + d) * NS + n]);
    h[n] = 0.f;
  }
  for (int c = 0; c < NCH; ++c) {
    const size_t off = (((size_t)b * KG + k) * NCH + c) * DI + d;
#pragma unroll
    for (int n = 0; n < NS; ++n) hs[off * NS + n] = h[n];
    const float S = sb[off];
#pragma unroll
    for (int n = 0; n < NS; ++n)
      h[n] = h[n] * __expf(A[n] * S) + qb[off * NS + n];
  }
}

// ---- 7) scan phase 3: re-run each chunk from h_start, emit y (+ D*u) -------
__global__ void k_scan_pass3(const float* __restrict__ delta,
                             const float* __restrict__ xc,
                             const float* __restrict__ xdbl,
                             const float* __restrict__ alog,
                             const float* __restrict__ hs,
                             const float* __restrict__ Ds,
                             float* __restrict__ oy) { // (B,K,L,192), scan order
  const int d = threadIdx.x;
  const int cid = blockIdx.x;
  const int c = cid % NCH, k = (cid / NCH) % KG, b = cid / (NCH * KG);
  float A[NS], h[NS];
  const size_t off = (((size_t)b * KG + k) * NCH + c) * DI + d;
#pragma unroll
  for (int n = 0; n < NS; ++n) {
    A[n] = -__expf(alog[((size_t)k * DI + d) * NS + n]);
    h[n] = hs[off * NS + n];
  }
  const float Dsk = Ds[k * DI + d];
  const size_t dbase  = ((size_t)b * KG + k) * LL * DI + d;
  const size_t xdbase = ((size_t)b * KG + k) * LL * CD;
  for (int t = 0; t < CL; ++t) {
    const int l = c * CL + t;
    const float dt = delta[dbase + (size_t)l * DI];
    const float u  = xc[((size_t)b * LL + sp_map(k, l)) * DI + d];
    const float du = dt * u;
    const float* Brow = xdbl + xdbase + (size_t)l * CD + RK;
    const float* Crow = Brow + NS;
    float y = 0.f;
#pragma unroll
    for (int n = 0; n < NS; ++n) {
      h[n] = h[n] * __expf(A[n] * dt) + du * Brow[n];
      y += h[n] * Crow[n];
    }
    oy[dbase + (size_t)l * DI] = y + Dsk * u;
  }
}

// ---- 8) merge 4 directions (index remap), LayerNorm over d, + z ------------
__global__ void k_combine_ln(const float* __restrict__ oy,
                             const float* __restrict__ zb,
                             const float* __restrict__ nw,
                             const float* __restrict__ nb,
                             float* __restrict__ yb) { // (B,L,192)
  const int d = threadIdx.x;
  const int bm = blockIdx.x;
  const int m = bm % LL, b = bm / LL;
  const int mt = (m % WW) * HH + (m / WW);  // wh<->hw transpose (involution)
  const int l0 = m, l1 = mt, l2 = LL - 1 - m, l3 = LL - 1 - mt;
  const size_t base = (size_t)b * KG * LL * DI;
  float y = oy[base + (size_t)l0 * DI + d]
          + oy[base + ((size_t)LL + l1) * DI + d]
          + oy[base + ((size_t)2 * LL + l2) * DI + d]
          + oy[base + ((size_t)3 * LL + l3) * DI + d];
  __shared__ float sA[DI], sB[DI];
  __shared__ float smu, srstd;
  sA[d] = y;
  sB[d] = y * y;
  __syncthreads();
  for (int s = 96; s >= 3; s >>= 1) {   // 96,48,24,12,6,3
    if (d < s) { sA[d] += sA[d + s]; sB[d] += sB[d + s]; }
    __syncthreads();
  }
  if (d == 0) {
    float s1 = sA[0] + sA[1] + sA[2];
    float s2 = sB[0] + sB[1] + sB[2];
    float mu = s1 / DI;
    smu = mu;
    srstd = rsqrtf(s2 / DI - mu * mu + 1e-5f);
  }
  __syncthreads();
  yb[((size_t)b * LL + m) * DI + d] =
      (y - smu) * srstd * nw[d] + nb[d] + zb[((size_t)b * LL + m) * DI + d];
}

// ---- 9) out_proj: out[b,c,hw] = sum_d yb[b,hw,d] * out_proj_w[c,d] ---------
__global__ void k_outproj(const float* __restrict__ yb,
                          const float* __restrict__ w,   // (96,192)
                          float* __restrict__ out) {     // (B,96,L)
  const int lane = threadIdx.x;
  const int half = lane >> 4, t16 = lane & 15;
  const int koff = 2 * half;
  const int m0 = blockIdx.x * 16;
  const int n0 = blockIdx.y * 16;
  const float* arow = yb + (size_t)(m0 + t16) * DI;
  const float* brow = w + (size_t)(n0 + t16) * DI;
  v8f c = wave_tile_gemm(arow, brow, DI, koff, true);
  const int col = n0 + t16;
#pragma unroll
  for (int r = 0; r < 8; ++r) {
    int g = m0 + r + 8 * half;
    int b = g / LL, l = g % LL;
    out[((size_t)b * DM + col) * LL + l] = c[r];
  }
}

extern "C" void kernel_launch(void* const* d_in, const int* in_sizes, int n_in,
                              void* d_out, int out_size, void* d_ws, size_t ws_size,
                              hipStream_t stream) {
  (void)in_sizes; (void)n_in; (void)out_size; (void)ws_size;
  const float* x    = (const float*)d_in[0];
  const float* inw  = (const float*)d_in[1];
  const float* cw   = (const float*)d_in[2];
  const float* cb   = (const float*)d_in[3];
  const float* xpw  = (const float*)d_in[4];
  const float* dtw  = (const float*)d_in[5];
  const float* dtb  = (const float*)d_in[6];
  const float* alog = (const float*)d_in[7];
  const float* Ds   = (const float*)d_in[8];
  const float* nw   = (const float*)d_in[9];
  const float* nb   = (const float*)d_in[10];
  const float* opw  = (const float*)d_in[11];
  float* out = (float*)d_out;

  float* ws = (float*)d_ws;
  size_t o = 0;
  float* xp    = ws + o; o += (size_t)BN * LL * DI;
  float* zb    = ws + o; o += (size_t)BN * LL * DI;
  float* xc    = ws + o; o += (size_t)BN * LL * DI;
  float* xdbl  = ws + o; o += (size_t)BN * KG * LL * CD;
  float* delta = ws + o; o += (size_t)BN * KG * LL * DI;
  float* qb    = ws + o; o += (size_t)BN * KG * NCH * DI * NS;
  float* sb    = ws + o; o += (size_t)BN * KG * NCH * DI;
  float* hs    = ws + o; o += (size_t)BN * KG * NCH * DI * NS;
  float* oy    = ws + o; o += (size_t)BN * KG * LL * DI;
  float* yb    = ws + o; o += (size_t)BN * LL * DI;

  k_inproj<<<dim3(BN * LL / 16, (2 * DI) / 16), dim3(32), 0, stream>>>(x, inw, xp, zb);
  {
    int n = BN * LL * DI;
    k_conv_silu<<<dim3((n + 255) / 256), dim3(256), 0, stream>>>(xp, cw, cb, xc);
  }
  k_xdbl<<<dim3(LL / 16, 3, BN * KG), dim3(32), 0, stream>>>(xc, xpw, xdbl);
  {
    int n = BN * KG * LL * DI;
    k_delta<<<dim3((n + 255) / 256), dim3(256), 0, stream>>>(xdbl, dtw, dtb, delta);
  }
  k_scan_pass1<<<dim3(BN * KG * NCH), dim3(DI), 0, stream>>>(delta, xc, xdbl, alog, qb, sb);
  k_scan_pass2<<<dim3(BN * KG), dim3(DI), 0, stream>>>(qb, sb, alog, hs);
  k_scan_pass3<<<dim3(BN * KG * NCH), dim3(DI), 0, stream>>>(delta, xc, xdbl, alog, hs, Ds, oy);
  k_combine_ln<<<dim3(BN * LL), dim3(DI), 0, stream>>>(oy, zb, nw, nb, yb);
  k_outproj<<<dim3(BN * LL / 16, DM / 16), dim3(32), 0, stream>>>(yb, opw, out);
}